// garGCNConv_52871047413952
// MI455X (gfx1250) — compile-verified
//
#include <hip/hip_runtime.h>

typedef float v2f __attribute__((ext_vector_type(2)));
typedef float v4f __attribute__((ext_vector_type(4)));
typedef float v8f __attribute__((ext_vector_type(8)));

#define IN_C 128
#define OUT_C 32
#define LDSTRIDE 132         // 128 + 4 dword pad -> conflict-free b64 LDS frag reads
#define ROWS_PER_BLOCK 64    // 4 waves x 16-row WMMA tiles
#define GEMM_THREADS 128

// ---------------------------------------------------------------------------
// Kernel 1: h = x @ W^T + b   via V_WMMA_F32_16X16X4_F32 (exact fp32 on the
// matrix pipe; GEMM is bandwidth-trivial so no need to downcast).
// ---------------------------------------------------------------------------
__global__ __launch_bounds__(GEMM_THREADS)
void gcn_gemm_wmma(const float* __restrict__ x, const float* __restrict__ W,
                   const float* __restrict__ b, float* __restrict__ h, int nRows)
{
    __shared__ float xs[ROWS_PER_BLOCK * LDSTRIDE];   // 64 x 128 tile of x
    __shared__ float wls[OUT_C * LDSTRIDE];           // 32 x 128 W

    const int tid = threadIdx.x;
    const int blockRow0 = blockIdx.x * ROWS_PER_BLOCK;
    const int nv4row = IN_C / 4;                      // 32 float4 per row

    // ---- coalesced b128 staging: x tile -> LDS
    {
        const v4f* x4 = (const v4f*)x;
        #pragma unroll
        for (int i = 0; i < (ROWS_PER_BLOCK * nv4row) / GEMM_THREADS; ++i) {
            int j = i * GEMM_THREADS + tid;
            int r = j / nv4row, c4 = j % nv4row;
            v4f v = {0.f, 0.f, 0.f, 0.f};
            int gr = blockRow0 + r;
            if (gr < nRows) v = x4[(size_t)gr * nv4row + c4];
            *(v4f*)&xs[r * LDSTRIDE + c4 * 4] = v;
        }
        const v4f* W4 = (const v4f*)W;
        #pragma unroll
        for (int i = 0; i < (OUT_C * nv4row) / GEMM_THREADS; ++i) {
            int j = i * GEMM_THREADS + tid;
            int r = j / nv4row, c4 = j % nv4row;
            *(v4f*)&wls[r * LDSTRIDE + c4 * 4] = W4[r * nv4row + c4];
        }
    }
    __syncthreads();

    const int wave = tid >> 5;       // wave32
    const int lane = tid & 31;
    const int half = lane >> 4;      // K-half for A/B frags, M-half for C/D
    const int mn   = lane & 15;      // M index (A frags) / N index (B,C,D)

    const int tileRow0 = blockRow0 + wave * 16;
    if (tileRow0 >= nRows) return;   // N_NODES % 16 == 0 -> tiles all-or-nothing

    v8f c0 = {};                     // N columns 0..15
    v8f c1 = {};                     // N columns 16..31

    // ISA 32-bit A(16x4) layout: lane(half,m) holds {A[m][2h], A[m][2h+1]}
    // ISA 32-bit B(4x16) layout: lane(half,n) holds {B[2h][n], B[2h+1][n]} = W row pair
    const float* aBase  = &xs[(wave * 16 + mn) * LDSTRIDE + 2 * half];
    const float* b0Base = &wls[mn * LDSTRIDE + 2 * half];
    const float* b1Base = b0Base + 16 * LDSTRIDE;

    #pragma unroll
    for (int kk = 0; kk < IN_C / 4; ++kk) {
        v2f a  = *(const v2f*)(aBase  + kk * 4);
        v2f b0 = *(const v2f*)(b0Base + kk * 4);
        v2f b1 = *(const v2f*)(b1Base + kk * 4);
        c0 = __builtin_amdgcn_wmma_f32_16x16x4_f32(false, a, false, b0,
                                                   (short)0, c0, false, false);
        c1 = __builtin_amdgcn_wmma_f32_16x16x4_f32(false, a, false, b1,
                                                   (short)0, c1, false, false);
    }

    const float bias0 = b[mn];
    const float bias1 = b[16 + mn];
    // C/D layout: VGPR v, lane(half,n) = D[v + 8*half][n]
    #pragma unroll
    for (int v = 0; v < 8; ++v) {
        int row = tileRow0 + v + 8 * half;
        h[(size_t)row * OUT_C + mn]      = c0[v] + bias0;
        h[(size_t)row * OUT_C + 16 + mn] = c1[v] + bias1;
    }
}

// ---------------------------------------------------------------------------
// Kernel 2: out[n] = sum_e w_e * h[src_e] + self_w[n] * h[n]
// One wave per node; 4 edges x 8 float4 channel-groups per iteration so each
// gather is a full-wave 512B b128 burst out of (L2-resident) h.
// ---------------------------------------------------------------------------
__global__ __launch_bounds__(256)
void gcn_aggregate(const float* __restrict__ h, const float* __restrict__ ew,
                   const float* __restrict__ sw, const int* __restrict__ tar_ptr,
                   const int* __restrict__ src, float* __restrict__ out, int nNodes)
{
    const int lane = threadIdx.x & 31;
    const int node = (int)((blockIdx.x * blockDim.x + threadIdx.x) >> 5);
    if (node >= nNodes) return;

    const int start = tar_ptr[node];
    const int end   = tar_ptr[node + 1];

    const int c8   = lane & 7;       // channel quad: floats c8*4 .. c8*4+3
    const int slot = lane >> 3;      // edge slot 0..3

    const v4f* h4 = (const v4f*)h;
    v4f acc = {0.f, 0.f, 0.f, 0.f};

    for (int e0 = start; e0 < end; e0 += 4) {
        int e = e0 + slot;
        if (e < end) {
            int   s = src[e];
            float w = ew[e];
            v4f hv = h4[(size_t)s * (OUT_C / 4) + c8];
            v4f wv = {w, w, w, w};
            acc += wv * hv;
        }
    }

    // reduce the 4 edge slots: lanes {c8, c8+8, c8+16, c8+24}
    #pragma unroll
    for (int i = 0; i < 4; ++i) {
        acc[i] += __shfl_xor(acc[i], 8, 32);
        acc[i] += __shfl_xor(acc[i], 16, 32);
    }

    if (slot == 0) {
        float s = sw[node];
        v4f sv = {s, s, s, s};
        v4f hs = h4[(size_t)node * (OUT_C / 4) + c8];
        v4f o  = acc + sv * hs;
        *(v4f*)&out[(size_t)node * OUT_C + c8 * 4] = o;
    }
}

// ---------------------------------------------------------------------------
extern "C" void kernel_launch(void* const* d_in, const int* in_sizes, int n_in,
                              void* d_out, int out_size, void* d_ws, size_t ws_size,
                              hipStream_t stream) {
    const float* x   = (const float*)d_in[0];
    const float* W   = (const float*)d_in[1];
    const float* b   = (const float*)d_in[2];
    const float* ew  = (const float*)d_in[3];
    const float* sw  = (const float*)d_in[4];
    const int*   tp  = (const int*)d_in[5];
    const int*   src = (const int*)d_in[6];
    float*       out = (float*)d_out;

    const int nNodes = in_sizes[4];          // self_edge_weight has N_NODES elems
    float* h = (float*)d_ws;                 // N_NODES * OUT_C * 4 = 12.8 MB

    const int gemmBlocks = (nNodes + ROWS_PER_BLOCK - 1) / ROWS_PER_BLOCK;
    gcn_gemm_wmma<<<gemmBlocks, GEMM_THREADS, 0, stream>>>(x, W, b, h, nNodes);

    const int wavesPerBlock = 256 / 32;
    const int aggBlocks = (nNodes + wavesPerBlock - 1) / wavesPerBlock;
    gcn_aggregate<<<aggBlocks, 256, 0, stream>>>(h, ew, sw, tp, src, out, nNodes);
}